// VectorQuantizer_55198919688816
// MI455X (gfx1250) — compile-verified
//
#include <hip/hip_runtime.h>
#include <hip/hip_bf16.h>
#include <math.h>

// Problem constants
#define BN   64          // batch
#define DD   64          // embedding dim
#define HW   1024        // 32*32
#define NTOK 65536       // BN*HW tokens
#define KCB  1024        // codebook size
#define QELEMS 4194304   // BN*DD*HW

// d_out layout (floats)
#define OFF_LOSS 0
#define OFF_QUANT 1
#define OFF_PERP 4194305
#define OFF_ENC  4194306
#define OFF_IDX  71303170

// LDS layout for argmin kernel (floats):
//   codebook pair-interleaved: lds[(d>>1)*PPITCH + 2*k + (d&1)]
//   PPITCH = 2*KCB + 32 pad -> pitch % 64 == 32, so the hi half-wave (pair
//   offset) lands on banks 32..63 while lanes 0-15 use 0..31: conflict-free b64.
#define PPITCH   (2 * KCB + 32)            // 2080 floats per d-pair
#define NPAIRS   (DD / 2)                  // 32
#define NORM_OFF (NPAIRS * PPITCH)         // 66560
#define LDS_FLOATS (NORM_OFF + KCB)        // 67584
#define LDS_BYTES (LDS_FLOATS * 4)         // 270336 < 320KB

typedef __attribute__((ext_vector_type(2))) float v2f;
typedef __attribute__((ext_vector_type(8))) float v8f;

// ---------------------------------------------------------------------------
// Kernel 0: zero the small workspace accumulators (ws poisoned once, not reset)
__global__ __launch_bounds__(256) void vq_zero(int* ws_count) {
    int t = blockIdx.x * 256 + threadIdx.x;
    if (t < KCB) ws_count[t] = 0;
}

// ---------------------------------------------------------------------------
// Kernel 1: argmin over codebook via V_WMMA_F32_16X16X4_F32
// One wave handles 16 tokens; block = 8 waves; grid = NTOK/128 = 512 blocks.
__global__ __launch_bounds__(256) void vq_argmin(const float* __restrict__ z,
                                                 const float* __restrict__ emb,
                                                 int* __restrict__ ws_idx) {
    extern __shared__ float lds[];

    const int tid = threadIdx.x;

    // Stage codebook pair-interleaved: lds[(d>>1)*PPITCH + 2k + (d&1)] = emb[k,d]
    for (int i = tid; i < KCB * DD; i += 256) {
        int k = i >> 6;
        int d = i & 63;
        lds[(d >> 1) * PPITCH + 2 * k + (d & 1)] = emb[i];
    }
    __syncthreads();

    // Per-code squared norms (exact fp32)
    for (int k = tid * 4; k < tid * 4 + 4; ++k) {
        float s = 0.0f;
        #pragma unroll
        for (int d = 0; d < DD; ++d) {
            float e = lds[(d >> 1) * PPITCH + 2 * k + (d & 1)];
            s += e * e;
        }
        lds[NORM_OFF + k] = s;
    }
    __syncthreads();

    const int lane = tid & 31;
    const int wv   = tid >> 5;
    const int col  = lane & 15;     // N (or M) index within 16
    const int hi   = lane >> 4;     // 0: K=0,1 ; 1: K=2,3 (and rows 8..15 of C)

    const int m0 = (blockIdx.x * 8 + wv) * 16;   // first token of tile
    const int n  = m0 + col;
    const int bb = n >> 10;
    const int pp = n & 1023;
    const float* zb = z + (size_t)bb * (DD * HW) + pp;

    // A tile: 16 tokens x 64 dims, in 16 chunks of K=4 (16x4 f32 A layout):
    // lane<16 holds (K=4kc, 4kc+1) for M=lane; lane>=16 holds (4kc+2, 4kc+3).
    v2f A[16];
    #pragma unroll
    for (int kc = 0; kc < 16; ++kc) {
        int d = kc * 4 + hi * 2;
        A[kc].x = zb[(size_t)d * HW];
        A[kc].y = zb[(size_t)(d + 1) * HW];
    }

    float bestS[8];
    int   bestK[8];
    #pragma unroll
    for (int j = 0; j < 8; ++j) { bestS[j] = 3.4e38f; bestK[j] = 0; }

    for (int k0 = 0; k0 < KCB; k0 += 16) {
        // Two independent accumulator chains to break the WMMA RAW chain.
        v8f c0 = {0.f, 0.f, 0.f, 0.f, 0.f, 0.f, 0.f, 0.f};
        v8f c1 = {0.f, 0.f, 0.f, 0.f, 0.f, 0.f, 0.f, 0.f};
        #pragma unroll
        for (int kc = 0; kc < 16; kc += 2) {
            int pairA = kc * 2 + hi;          // (kc*4 + hi*2) >> 1
            int pairB = (kc + 1) * 2 + hi;
            v2f Bv0 = *(const v2f*)&lds[pairA * PPITCH + 2 * (k0 + col)];
            v2f Bv1 = *(const v2f*)&lds[pairB * PPITCH + 2 * (k0 + col)];
            c0 = __builtin_amdgcn_wmma_f32_16x16x4_f32(
                    false, A[kc],     false, Bv0, (short)0, c0, false, false);
            c1 = __builtin_amdgcn_wmma_f32_16x16x4_f32(
                    false, A[kc + 1], false, Bv1, (short)0, c1, false, false);
        }
        const int kk = k0 + col;
        const float nrm = lds[NORM_OFF + kk];
        #pragma unroll
        for (int j = 0; j < 8; ++j) {
            float s = fmaf(-2.0f, c0[j] + c1[j], nrm);  // ||e||^2 - 2 z.e
            if (s < bestS[j] || (s == bestS[j] && kk < bestK[j])) {
                bestS[j] = s; bestK[j] = kk;
            }
        }
    }

    // Reduce across the 16 columns (lanes 0-15 hold rows 0..7; 16-31 rows 8..15)
    #pragma unroll
    for (int j = 0; j < 8; ++j) {
        float s = bestS[j];
        int   k = bestK[j];
        #pragma unroll
        for (int off = 8; off >= 1; off >>= 1) {
            float s2 = __shfl_xor(s, off, 16);
            int   k2 = __shfl_xor(k, off, 16);
            if (s2 < s || (s2 == s && k2 < k)) { s = s2; k = k2; }
        }
        if (col == 0) ws_idx[m0 + hi * 8 + j] = k;
    }
}

// ---------------------------------------------------------------------------
// Kernel 2: one-hot encodings, 256 MB stream, float2 stores (8B-aligned offset)
__global__ __launch_bounds__(256) void vq_encodings(const int* __restrict__ ws_idx,
                                                    float* __restrict__ out_enc) {
    int gid = blockIdx.x * 256 + threadIdx.x;     // NTOK*KCB/2 threads
    int n  = gid >> 9;
    int k2 = (gid & 511) << 1;
    int k  = ws_idx[n];
    float2 v;
    v.x = (k == k2)     ? 1.0f : 0.0f;
    v.y = (k == k2 + 1) ? 1.0f : 0.0f;
    *(float2*)(out_enc + (size_t)gid * 2) = v;
}

// ---------------------------------------------------------------------------
// Kernel 3: quantized output (NCHW) + deterministic partial sums of (q-z)^2
__global__ __launch_bounds__(256) void vq_quant_loss(const float* __restrict__ z,
                                                     const float* __restrict__ emb,
                                                     const int* __restrict__ ws_idx,
                                                     float* __restrict__ out_quant,
                                                     float* __restrict__ partial) {
    int gid = blockIdx.x * 256 + threadIdx.x;     // QELEMS/4 threads
    int o0 = gid * 4;
    int p0 = o0 & 1023;
    int d  = (o0 >> 10) & 63;
    int b  = o0 >> 16;
    const float4 zv = *(const float4*)(z + o0);
    float zz[4] = {zv.x, zv.y, zv.z, zv.w};
    float acc = 0.0f;
    #pragma unroll
    for (int i = 0; i < 4; ++i) {
        int n = b * HW + p0 + i;
        int k = ws_idx[n];
        float q = emb[k * DD + d];
        out_quant[o0 + i] = q;                    // ST fwd value == quant
        float df = q - zz[i];
        acc += df * df;
    }
    // deterministic block reduction
    const int lane = threadIdx.x & 31;
    const int wv   = threadIdx.x >> 5;
    #pragma unroll
    for (int off = 16; off >= 1; off >>= 1) acc += __shfl_xor(acc, off, 32);
    __shared__ float wsum[8];
    if (lane == 0) wsum[wv] = acc;
    __syncthreads();
    if (wv == 0) {
        float s = (lane < 8) ? wsum[lane] : 0.0f;
        #pragma unroll
        for (int off = 16; off >= 1; off >>= 1) s += __shfl_xor(s, off, 32);
        if (lane == 0) partial[blockIdx.x] = s;
    }
}

// ---------------------------------------------------------------------------
// Kernel 4: code usage counts (exact int atomics) + idx written as float
__global__ __launch_bounds__(256) void vq_counts(const int* __restrict__ ws_idx,
                                                 int* __restrict__ ws_count,
                                                 float* __restrict__ out_idx) {
    int n = blockIdx.x * 256 + threadIdx.x;       // NTOK threads
    int k = ws_idx[n];
    atomicAdd(&ws_count[k], 1);
    out_idx[n] = (float)k;
}

// ---------------------------------------------------------------------------
// Kernel 5: finalize loss + perplexity (single block, fixed-order reduction)
__global__ __launch_bounds__(256) void vq_finalize(const float* __restrict__ partial,
                                                   int nPartial,
                                                   const int* __restrict__ ws_count,
                                                   float* __restrict__ d_out) {
    __shared__ float sh[256];
    int t = threadIdx.x;

    float s = 0.0f;
    for (int i = t; i < nPartial; i += 256) s += partial[i];
    sh[t] = s;
    __syncthreads();
    for (int off = 128; off >= 1; off >>= 1) {
        if (t < off) sh[t] += sh[t + off];
        __syncthreads();
    }
    if (t == 0) d_out[OFF_LOSS] = 1.25f * sh[0] / (float)QELEMS;
    __syncthreads();

    float e = 0.0f;
    for (int i = t; i < KCB; i += 256) {
        float pr = (float)ws_count[i] * (1.0f / (float)NTOK);
        e += pr * logf(pr + 1e-10f);
    }
    sh[t] = e;
    __syncthreads();
    for (int off = 128; off >= 1; off >>= 1) {
        if (t < off) sh[t] += sh[t + off];
        __syncthreads();
    }
    if (t == 0) d_out[OFF_PERP] = expf(-sh[0]);
}

// ---------------------------------------------------------------------------
extern "C" void kernel_launch(void* const* d_in, const int* in_sizes, int n_in,
                              void* d_out, int out_size, void* d_ws, size_t ws_size,
                              hipStream_t stream) {
    const float* z   = (const float*)d_in[0];   // [64,64,32,32]
    const float* emb = (const float*)d_in[1];   // [1024,64]
    float* out = (float*)d_out;

    int*   ws_idx   = (int*)d_ws;                                   // NTOK ints
    int*   ws_count = (int*)((char*)d_ws + (size_t)NTOK * 4);       // KCB ints
    float* ws_part  = (float*)((char*)d_ws + (size_t)(NTOK + KCB) * 4); // 4096 floats

    vq_zero<<<4, 256, 0, stream>>>(ws_count);

    vq_argmin<<<NTOK / 128, 256, LDS_BYTES, stream>>>(z, emb, ws_idx);

    vq_encodings<<<(NTOK * (KCB / 2)) / 256, 256, 0, stream>>>(ws_idx, out + OFF_ENC);

    vq_quant_loss<<<QELEMS / 1024, 256, 0, stream>>>(z, emb, ws_idx,
                                                     out + OFF_QUANT, ws_part);

    vq_counts<<<NTOK / 256, 256, 0, stream>>>(ws_idx, ws_count, out + OFF_IDX);

    vq_finalize<<<1, 256, 0, stream>>>(ws_part, QELEMS / 1024, ws_count, out);
}